// BaseValidationLoss_57690000720629
// MI455X (gfx1250) — compile-verified
//
#include <hip/hip_runtime.h>

// Problem constants (reference: RES=(480,640), FLOW_SCALING=128, MAX_TS=1)
#define H_RES 480
#define W_RES 640
#define HW_RES (H_RES * W_RES)
#define FLOW_SCALE 128.0f

constexpr int TILE = 256;          // events per tile == blockDim.x
constexpr int BLOCKS_PER_BATCH = 512;

// ---- CDNA5 async global->LDS copy helpers (ASYNCcnt-tracked) ----------------
__device__ __forceinline__ void async_lds_load_b128(unsigned lds_off, const void* g) {
    asm volatile("global_load_async_to_lds_b128 %0, %1, off"
                 :: "v"(lds_off), "v"((unsigned long long)(uintptr_t)g)
                 : "memory");
}
__device__ __forceinline__ void async_lds_load_b64(unsigned lds_off, const void* g) {
    asm volatile("global_load_async_to_lds_b64 %0, %1, off"
                 :: "v"(lds_off), "v"((unsigned long long)(uintptr_t)g)
                 : "memory");
}
__device__ __forceinline__ void wait_async_le0() {
    asm volatile("s_wait_asynccnt 0x0" ::: "memory");
}
__device__ __forceinline__ void wait_async_le2() {
    asm volatile("s_wait_asynccnt 0x2" ::: "memory");
}

// ---- main scatter kernel (placed FIRST so the disasm snippet shows it) ------
// ev:   [B, N, 4] (ts, y, x, p) as float4 per event
// pm:   [B, N, 2] polarity mask as float2 per event
// flow: [B, 2, H, W]  (channel 0 = x-flow, channel 1 = y-flow)
// out:  [B, 2, H, W]  (channel 0 = positive IWE, channel 1 = negative IWE)
__global__ __launch_bounds__(TILE)
void iwe_scatter_kernel(const float4* __restrict__ ev,
                        const float2* __restrict__ pm,
                        const float*  __restrict__ flow,
                        float*        __restrict__ out,
                        int nEvents) {
    __shared__ float4 sEv[2][TILE];
    __shared__ float2 sPm[2][TILE];

    const int b   = blockIdx.y;
    const int tid = threadIdx.x;

    const float4* evB = ev + (size_t)b * nEvents;
    const float2* pmB = pm + (size_t)b * nEvents;
    const float*  flB = flow + (size_t)b * (2 * HW_RES);
    float*        oPos = out + (size_t)b * (2 * HW_RES);
    float*        oNeg = oPos + HW_RES;

    const unsigned ldsEv0 = (unsigned)(uintptr_t)&sEv[0][tid];
    const unsigned ldsEv1 = (unsigned)(uintptr_t)&sEv[1][tid];
    const unsigned ldsPm0 = (unsigned)(uintptr_t)&sPm[0][tid];
    const unsigned ldsPm1 = (unsigned)(uintptr_t)&sPm[1][tid];

    const int tilesTotal = (nEvents + TILE - 1) / TILE;
    const int tile0 = blockIdx.x;

    // Prologue: prefetch first tile (clamped index keeps issue wave-uniform)
    if (tile0 < tilesTotal) {
        int i0 = tile0 * TILE + tid;
        if (i0 >= nEvents) i0 = nEvents - 1;
        async_lds_load_b128(ldsEv0, evB + i0);
        async_lds_load_b64 (ldsPm0, pmB + i0);
    }

    int cur = 0;
    for (int t = tile0; t < tilesTotal; t += gridDim.x) {
        const int tn = t + (int)gridDim.x;
        if (tn < tilesTotal) {
            // Issue next tile into the other buffer, then wait for the
            // current tile. At most 4 loads outstanding (2 current + 2 next);
            // loads complete in order, so <=2 outstanding => current done.
            int in_ = tn * TILE + tid;
            if (in_ >= nEvents) in_ = nEvents - 1;
            async_lds_load_b128(cur ? ldsEv0 : ldsEv1, evB + in_);
            async_lds_load_b64 (cur ? ldsPm0 : ldsPm1, pmB + in_);
            wait_async_le2();
        } else {
            wait_async_le0();
        }

        const int idx = t * TILE + tid;
        if (idx < nEvents) {
            const float4 e   = sEv[cur][tid];   // (ts, y, x, p)
            const float2 pmv = sPm[cur][tid];   // (pos, neg)

            // per-event flow gather at the event's integer pixel
            const int lin0 = (int)e.y * W_RES + (int)e.z;
            const float fx = flB[lin0];            // channel 0: x-flow
            const float fy = flB[HW_RES + lin0];   // channel 1: y-flow

            // warp: pos + (MAX_TS - ts) * flow * FLOW_SCALING
            const float dt = (1.0f - e.x) * FLOW_SCALE;
            const float wy = fmaf(dt, fy, e.y);
            const float wx = fmaf(dt, fx, e.z);

            const float tyf = floorf(wy), lxf = floorf(wx);
            const float ay = wy - tyf;   // weight toward by = ty+1
            const float ax = wx - lxf;   // weight toward rx = lx+1
            const float byf = tyf + 1.0f, rxf = lxf + 1.0f;

            const float w00 = (1.0f - ay) * (1.0f - ax); // (ty,lx)
            const float w01 = (1.0f - ay) * ax;          // (ty,rx)
            const float w10 = ay * (1.0f - ax);          // (by,lx)
            const float w11 = ay * ax;                   // (by,rx)

            const bool tyIn = (tyf >= 0.0f) && (tyf < (float)H_RES);
            const bool byIn = (byf >= 0.0f) && (byf < (float)H_RES);
            const bool lxIn = (lxf >= 0.0f) && (lxf < (float)W_RES);
            const bool rxIn = (rxf >= 0.0f) && (rxf < (float)W_RES);

            const int ity = (int)tyf;
            const int ilx = (int)lxf;
            const float p0 = pmv.x, p1 = pmv.y;

            // 4 corners; OOB corners have masked weight == 0 in the
            // reference, so skipping them is exact.
            #define SPLAT(OK, YY, XX, WW)                                        \
                if (OK) {                                                        \
                    const int l = (YY) * W_RES + (XX);                           \
                    if (p0 != 0.0f) unsafeAtomicAdd(oPos + l, (WW) * p0);        \
                    if (p1 != 0.0f) unsafeAtomicAdd(oNeg + l, (WW) * p1);        \
                }
            SPLAT(tyIn && lxIn, ity,     ilx,     w00)
            SPLAT(tyIn && rxIn, ity,     ilx + 1, w01)
            SPLAT(byIn && lxIn, ity + 1, ilx,     w10)
            SPLAT(byIn && rxIn, ity + 1, ilx + 1, w11)
            #undef SPLAT
        }
        cur ^= 1;
    }
}

// ---- zero-fill output (harness does not re-poison between replays) ----------
__global__ void iwe_zero_kernel(float4* __restrict__ p4, int n4,
                                float* __restrict__ p, int n) {
    const int i = blockIdx.x * blockDim.x + threadIdx.x;
    if (i < n4) p4[i] = make_float4(0.f, 0.f, 0.f, 0.f);
    if (i == 0) {                       // scalar tail (dead for our sizes)
        for (int k = n4 * 4; k < n; ++k) p[k] = 0.f;
    }
}

extern "C" void kernel_launch(void* const* d_in, const int* in_sizes, int n_in,
                              void* d_out, int out_size, void* d_ws, size_t ws_size,
                              hipStream_t stream) {
    (void)n_in; (void)d_ws; (void)ws_size;

    const float4* ev   = reinterpret_cast<const float4*>(d_in[0]); // [B,N,4]
    const float*  flow = reinterpret_cast<const float*>(d_in[1]);  // [B,2,H,W]
    const float2* pm   = reinterpret_cast<const float2*>(d_in[2]); // [B,N,2]
    float* out = reinterpret_cast<float*>(d_out);                  // [B,2,H,W]

    const int B = in_sizes[1] / (2 * HW_RES);
    const int N = in_sizes[0] / (4 * B);

    // 1) zero the accumulator every call
    {
        const int n4 = out_size / 4;
        const int thr = 256;
        iwe_zero_kernel<<<(n4 + thr - 1) / thr, thr, 0, stream>>>(
            reinterpret_cast<float4*>(out), n4, out, out_size);
    }

    // 2) async-pipelined gather/warp/scatter
    dim3 grid(BLOCKS_PER_BATCH, B);
    iwe_scatter_kernel<<<grid, TILE, 0, stream>>>(ev, pm, flow, out, N);
}